// PolicyNet_18098992185849
// MI455X (gfx1250) — compile-verified
//
#include <hip/hip_runtime.h>
#include <math.h>
#include <stdint.h>

// ---------------------------------------------------------------------------
// PolicyNet pipeline for MI455X (gfx1250, wave32).
//  - Dense GEMMs: V_WMMA_F32_16X16X4_F32, one 16x32 C tile per wave
//    (2 accumulators, A fragment reused across both N tiles).
//  - Attention: flash-style streaming softmax; K/V tiles staged to LDS via
//    GLOBAL_LOAD_ASYNC_TO_LDS_B32 (ASYNCcnt) when the builtin is available.
//  - Sinkhorn: 200 graph-captured sweeps; row sweep reads cost row-major,
//    col sweep reads a transposed copy so both directions are coalesced
//    (~1.7GB L2-resident traffic -> ~73us at 23.3TB/s).
// ---------------------------------------------------------------------------

typedef __attribute__((ext_vector_type(2))) float v2f;
typedef __attribute__((ext_vector_type(8))) float v8f;

static constexpr int N_NODES = 1024;
static constexpr int N_EDGES = 8192;
static constexpr int N_DEV   = 2048;
static constexpr int DIM     = 96;
static constexpr int NHEAD   = 4;
static constexpr int DHD     = 24;     // DIM / NHEAD
static constexpr int FFDIM   = 256;
static constexpr int NLAYERS = 4;
static constexpr int MROW    = 1024;   // cost rows (no bin)
static constexpr int NCOL    = 2048;
static constexpr int SINK_IT = 100;
static constexpr float NRM   = 6.907755278982137f;   // -log(1e-3)
static constexpr float NEGBIG = -3.4e38f;

#if defined(__gfx1250__) && __has_builtin(__builtin_amdgcn_global_load_async_to_lds_b32)
#define ASYNC_LDS 1
#endif

typedef __attribute__((address_space(1))) int g1_int;
typedef __attribute__((address_space(3))) int l3_int;

__device__ __forceinline__ void async_g2l_b32(const float* g, float* l) {
#ifdef ASYNC_LDS
    __builtin_amdgcn_global_load_async_to_lds_b32(
        (g1_int*)(uintptr_t)g,
        (l3_int*)(uint32_t)(uintptr_t)l,
        0, 0);
#else
    *l = *g;
#endif
}

__device__ __forceinline__ void wait_async_lds() {
#ifdef ASYNC_LDS
#if __has_builtin(__builtin_amdgcn_s_wait_asynccnt)
    __builtin_amdgcn_s_wait_asynccnt(0);
#else
    asm volatile("s_wait_asynccnt 0x0" ::: "memory");
#endif
#endif
}

// -------------------------- elementwise helpers ----------------------------

__global__ void zero_kernel(float* p, int n) {
    int i = blockIdx.x * blockDim.x + threadIdx.x;
    if (i < n) p[i] = 0.f;
}

__global__ void copy_kernel(const float* __restrict__ a, float* __restrict__ b, int n) {
    int i = blockIdx.x * blockDim.x + threadIdx.x;
    if (i < n) b[i] = a[i];
}

__global__ void deg_kernel(const int* __restrict__ dst, float* deg, int nE) {
    int i = blockIdx.x * blockDim.x + threadIdx.x;
    if (i < nE) atomicAdd(&deg[dst[i]], 1.f);
}

__global__ void norm_kernel(const float* __restrict__ deg, float* __restrict__ nrm, int n) {
    int i = blockIdx.x * blockDim.x + threadIdx.x;
    if (i < n) {
        float d = deg[i] + 1.f;            // + self loop
        nrm[i] = rsqrtf(fmaxf(d, 1.f));
    }
}

__global__ void rowscale_kernel(const float* __restrict__ h, const float* __restrict__ nrm,
                                float* __restrict__ out, int n, int F) {
    int i = blockIdx.x * blockDim.x + threadIdx.x;
    if (i < n * F) out[i] = h[i] * nrm[i / F];
}

__global__ void scatter_kernel(const float* __restrict__ hs, const int* __restrict__ src,
                               const int* __restrict__ dst, float* acc, int F) {
    int t = blockIdx.x * blockDim.x + threadIdx.x;
    if (t < N_EDGES * F) {
        int e = t / F, f = t % F;
        atomicAdd(&acc[dst[e] * F + f], hs[src[e] * F + f]);
    }
}

// acc = (acc + self-loop contribution) * norm[row]
__global__ void combine_kernel(float* __restrict__ acc, const float* __restrict__ hs,
                               const float* __restrict__ nrm, int n, int F) {
    int i = blockIdx.x * blockDim.x + threadIdx.x;
    if (i < n * F) acc[i] = (acc[i] + hs[i]) * nrm[i / F];
}

// ------------------------------- WMMA GEMM ---------------------------------
// C[M x N] = act(A[M x lda] @ W[N x ldw]^T + bias) (+ resid).  One 16x32 C
// tile per wave32 (two f32 accumulators sharing the A fragment); K stepped
// by 4 with V_WMMA_F32_16X16X4_F32.
// A layout (16x4, ISA 7.12.2): lanes 0-15 M=lane {v0=K0, v1=K1};
//                              lanes 16-31 M=lane-16 {v0=K2, v1=K3}.
// B layout (4x16): lanes hold N=lane&15; v0 = row K0|K2, v1 = row K1|K3.
// C/D layout: VGPR r -> M = r (+8 for upper half-wave), N = lane&15.

template <bool RELU>
__global__ void gemm_wmma_kernel(const float* __restrict__ A, int lda,
                                 const float* __restrict__ W, int ldw,
                                 const float* __restrict__ bias,
                                 const float* __restrict__ resid,
                                 float* __restrict__ C, int ldc,
                                 int M, int N, int K) {
    int wave = threadIdx.x >> 5;
    int lane = threadIdx.x & 31;
    int tilesN = N >> 5;                          // 32-wide tiles
    int nTiles = (M >> 4) * tilesN;
    int tileId = blockIdx.x * (blockDim.x >> 5) + wave;
    if (tileId >= nTiles) return;                 // wave-uniform: EXEC stays all-1s
    int tm = (tileId / tilesN) << 4;
    int tn = (tileId % tilesN) << 5;

    int half = lane >> 4;                         // 0: K0/K1, 1: K2/K3
    int l15  = lane & 15;
    const float* Arow = A + (size_t)(tm + l15) * lda;
    const float* W0   = W + (size_t)(tn + l15) * ldw;        // B[k][n] = W[n][k]
    const float* W1   = W + (size_t)(tn + 16 + l15) * ldw;

    v8f acc0 = {}, acc1 = {};
    for (int k = 0; k < K; k += 4) {
        int k0 = k + half * 2;
        v2f a, b0, b1;
        a.x  = Arow[k0]; a.y  = Arow[k0 + 1];
        b0.x = W0[k0];   b0.y = W0[k0 + 1];
        b1.x = W1[k0];   b1.y = W1[k0 + 1];
        acc0 = __builtin_amdgcn_wmma_f32_16x16x4_f32(
                   false, a, false, b0, (short)0, acc0, false, false);
        acc1 = __builtin_amdgcn_wmma_f32_16x16x4_f32(
                   false, a, false, b1, (short)0, acc1, false, false);
    }

    int col0 = tn + l15, col1 = tn + 16 + l15;
    float bv0 = bias ? bias[col0] : 0.f;
    float bv1 = bias ? bias[col1] : 0.f;
    #pragma unroll
    for (int r = 0; r < 8; ++r) {
        int row = tm + r + half * 8;
        float v0 = acc0[r] + bv0;
        float v1 = acc1[r] + bv1;
        if (resid) {
            v0 += resid[(size_t)row * ldc + col0];
            v1 += resid[(size_t)row * ldc + col1];
        }
        if (RELU) { v0 = fmaxf(v0, 0.f); v1 = fmaxf(v1, 0.f); }
        C[(size_t)row * ldc + col0] = v0;
        C[(size_t)row * ldc + col1] = v1;
    }
}

// --------------------------- streaming attention ---------------------------
// One thread per query row per head; K/V tiles of 128 keys staged in LDS via
// async global->LDS copies; online softmax (dh = 24).

__global__ void attn_kernel(const float* __restrict__ Q, int ldq,
                            const float* __restrict__ K, int ldk,
                            const float* __restrict__ V, int ldv,
                            float* __restrict__ O, int ldo, int nk) {
    __shared__ float sK[128 * DHD];
    __shared__ float sV[128 * DHD];
    int head = blockIdx.y;
    int q = blockIdx.x * blockDim.x + threadIdx.x;

    float qv[DHD];
    const float* qp = Q + (size_t)q * ldq + head * DHD;
    #pragma unroll
    for (int d = 0; d < DHD; ++d) qv[d] = qp[d];

    float m = NEGBIG, s = 0.f;
    float acc[DHD];
    #pragma unroll
    for (int d = 0; d < DHD; ++d) acc[d] = 0.f;
    const float scale = 0.2041241452319315f;      // 1/sqrt(24)

    for (int kb = 0; kb < nk; kb += 128) {
        for (int i = threadIdx.x; i < 128 * DHD; i += blockDim.x) {
            int kr = i / DHD, kd = i % DHD;
            async_g2l_b32(&K[(size_t)(kb + kr) * ldk + head * DHD + kd], &sK[i]);
            async_g2l_b32(&V[(size_t)(kb + kr) * ldv + head * DHD + kd], &sV[i]);
        }
        wait_async_lds();
        __syncthreads();
        for (int j = 0; j < 128; ++j) {
            float dot = 0.f;
            #pragma unroll
            for (int d = 0; d < DHD; ++d) dot += qv[d] * sK[j * DHD + d];
            float logit = dot * scale;
            float mn = fmaxf(m, logit);
            float c = __expf(m - mn);
            float p = __expf(logit - mn);
            s = s * c + p;
            #pragma unroll
            for (int d = 0; d < DHD; ++d) acc[d] = acc[d] * c + p * sV[j * DHD + d];
            m = mn;
        }
        __syncthreads();
    }
    float inv = 1.f / s;
    float* op = O + (size_t)q * ldo + head * DHD;
    #pragma unroll
    for (int d = 0; d < DHD; ++d) op[d] = acc[d] * inv;
}

// ------------------------------- layernorm ---------------------------------

__global__ void ln_kernel(const float* __restrict__ in, const float* __restrict__ g,
                          const float* __restrict__ b, float* __restrict__ out,
                          int n, int F) {
    int r = blockIdx.x * blockDim.x + threadIdx.x;
    if (r >= n) return;
    const float* x = in + (size_t)r * F;
    float mu = 0.f;
    for (int f = 0; f < F; ++f) mu += x[f];
    mu /= F;
    float var = 0.f;
    for (int f = 0; f < F; ++f) { float d = x[f] - mu; var += d * d; }
    var /= F;
    float rs = rsqrtf(var + 1e-5f);
    float* o = out + (size_t)r * F;
    for (int f = 0; f < F; ++f) o[f] = (x[f] - mu) * rs * g[f] + b[f];
}

// ---------------------------- cost / sinkhorn ------------------------------

__global__ void cost_kernel(const float* __restrict__ cg, const float* __restrict__ mt,
                            const float* __restrict__ binp, float* __restrict__ cost,
                            float* __restrict__ costT) {
    int t = blockIdx.x * blockDim.x + threadIdx.x;
    if (t >= (MROW + 1) * NCOL) return;
    int i = t / NCOL, j = t % NCOL;
    float val;
    if (i < MROW) {
        const float* a = cg + (size_t)i * DIM;
        const float* b = mt + (size_t)j * DIM;
        float sum = 0.f;
        #pragma unroll 8
        for (int k = 0; k < DIM; ++k) sum += fabsf(a[k] - b[k]);
        val = sum;
    } else {
        val = binp[0];
    }
    cost[t] = val;
    costT[(size_t)j * (MROW + 1) + i] = val;
}

__device__ __forceinline__ void lse_merge(float& m, float& s, float m2, float s2) {
    float mn = fmaxf(m, m2);
    s = s * __expf(m - mn) + s2 * __expf(m2 - mn);
    m = mn;
}

__global__ void sink_row_kernel(const float* __restrict__ cost, const float* __restrict__ v,
                                float* __restrict__ u) {
    int i = blockIdx.x;                            // 0 .. MROW (inclusive bin row)
    int tid = threadIdx.x;
    const float* row = cost + (size_t)i * NCOL;
    float m = NEGBIG, s = 0.f;
    for (int j = tid; j < NCOL; j += blockDim.x) {
        float val = row[j] + v[j];
        float mn = fmaxf(m, val);
        s = s * __expf(m - mn) + __expf(val - mn);
        m = mn;
    }
    __shared__ float sm[256], ss[256];
    sm[tid] = m; ss[tid] = s;
    __syncthreads();
    for (int o = 128; o > 0; o >>= 1) {
        if (tid < o) lse_merge(sm[tid], ss[tid], sm[tid + o], ss[tid + o]);
        __syncthreads();
    }
    if (tid == 0) {
        float lse = logf(ss[0]) + sm[0];
        float lmu = (i < MROW) ? NRM : (logf((float)(NCOL - MROW)) + NRM);
        u[i] = lmu - lse;
    }
}

__global__ void sink_col_kernel(const float* __restrict__ costT, const float* __restrict__ u,
                                float* __restrict__ v) {
    int j = blockIdx.x;                            // 0 .. NCOL-1
    int tid = threadIdx.x;
    const float* col = costT + (size_t)j * (MROW + 1);
    float m = NEGBIG, s = 0.f;
    for (int i = tid; i < MROW + 1; i += blockDim.x) {
        float val = col[i] + u[i];
        float mn = fmaxf(m, val);
        s = s * __expf(m - mn) + __expf(val - mn);
        m = mn;
    }
    __shared__ float sm[256], ss[256];
    sm[tid] = m; ss[tid] = s;
    __syncthreads();
    for (int o = 128; o > 0; o >>= 1) {
        if (tid < o) lse_merge(sm[tid], ss[tid], sm[tid + o], ss[tid + o]);
        __syncthreads();
    }
    if (tid == 0) v[j] = NRM - (logf(ss[0]) + sm[0]);
}

__global__ void scores_kernel(const float* __restrict__ cost, const float* __restrict__ u,
                              const float* __restrict__ v, float* __restrict__ out) {
    int t = blockIdx.x * blockDim.x + threadIdx.x;
    if (t >= (MROW + 1) * NCOL) return;
    int i = t / NCOL, j = t % NCOL;
    out[t] = __expf(cost[t] + u[i] + v[j] - NRM);
}

// ------------------------------ greedy pick --------------------------------

__global__ void greedy_kernel(const float* __restrict__ s0, float* __restrict__ choices,
                              float* __restrict__ chsc) {
    __shared__ float bv[1024];
    __shared__ int   bc[1024];
    __shared__ unsigned char used[NCOL];
    int t = threadIdx.x;
    used[t] = 0; used[t + 1024] = 0;
    __syncthreads();
    for (int r = 0; r < MROW; ++r) {
        float best = NEGBIG; int bestc = -1;
        for (int c = t; c < NCOL; c += 1024) {
            if (!used[c]) {
                float vv = s0[(size_t)r * NCOL + c];
                if (vv > best) { best = vv; bestc = c; }
            }
        }
        bv[t] = best; bc[t] = bestc;
        __syncthreads();
        for (int o = 512; o > 0; o >>= 1) {
            if (t < o) {
                float v2 = bv[t + o]; int c2 = bc[t + o];
                bool take = (v2 > bv[t]) ||
                            (v2 == bv[t] && c2 >= 0 && (bc[t] < 0 || c2 < bc[t]));
                if (take) { bv[t] = v2; bc[t] = c2; }
            }
            __syncthreads();
        }
        if (t == 0) {
            int c = bc[0];
            choices[r] = (float)c;
            chsc[r]    = (c >= 0) ? s0[(size_t)r * NCOL + c] : 0.f;
            if (c >= 0) used[c] = 1;
        }
        __syncthreads();
    }
}

// ------------------------------- launcher ----------------------------------

static inline int ceil_div(int a, int b) { return (a + b - 1) / b; }

static void launch_gemm(const float* A, int lda, const float* W, int ldw,
                        const float* bias, const float* resid,
                        float* C, int ldc, int M, int N, int K,
                        bool relu, hipStream_t s) {
    int tiles = (M / 16) * (N / 32);
    int blocks = ceil_div(tiles, 4);                 // 4 waves / block
    if (relu)
        hipLaunchKernelGGL((gemm_wmma_kernel<true>),  dim3(blocks), dim3(128), 0, s,
                           A, lda, W, ldw, bias, resid, C, ldc, M, N, K);
    else
        hipLaunchKernelGGL((gemm_wmma_kernel<false>), dim3(blocks), dim3(128), 0, s,
                           A, lda, W, ldw, bias, resid, C, ldc, M, N, K);
}

extern "C" void kernel_launch(void* const* d_in, const int* in_sizes, int n_in,
                              void* d_out, int out_size, void* d_ws, size_t ws_size,
                              hipStream_t stream) {
    const float* node_feat    = (const float*)d_in[0];
    const float* device_feats = (const float*)d_in[1];
    const float* sg_w[4] = {(const float*)d_in[2], (const float*)d_in[4],
                            (const float*)d_in[6], (const float*)d_in[8]};
    const float* sg_b[4] = {(const float*)d_in[3], (const float*)d_in[5],
                            (const float*)d_in[7], (const float*)d_in[9]};
    const float* sa_w  = (const float*)d_in[10];
    const float* sa_b  = (const float*)d_in[11];
    const float* sa_ow = (const float*)d_in[12];
    const float* sa_ob = (const float*)d_in[13];
    const float* ca_w  = (const float*)d_in[14];
    const float* ca_b  = (const float*)d_in[15];
    const float* ca_ow = (const float*)d_in[16];
    const float* ca_ob = (const float*)d_in[17];
    const float* ff_w1 = (const float*)d_in[18];
    const float* ff_b1 = (const float*)d_in[19];
    const float* ff_w2 = (const float*)d_in[20];
    const float* ff_b2 = (const float*)d_in[21];
    const float* ln1_g = (const float*)d_in[22];
    const float* ln1_b = (const float*)d_in[23];
    const float* ln2_g = (const float*)d_in[24];
    const float* ln2_b = (const float*)d_in[25];
    const float* ln3_g = (const float*)d_in[26];
    const float* ln3_b = (const float*)d_in[27];
    const float* match_w = (const float*)d_in[28];
    const float* match_b = (const float*)d_in[29];
    const float* bin_sc  = (const float*)d_in[30];
    const int*   edge_src = (const int*)d_in[31];
    const int*   edge_dst = (const int*)d_in[32];

    // workspace layout (floats)
    float* ws = (float*)d_ws;
    size_t off = 0;
    float* deg   = ws + off; off += N_NODES;
    float* nrm   = ws + off; off += N_NODES;
    float* bufA  = ws + off; off += (size_t)N_NODES * DIM;   // graph conv h / cg
    float* bufB  = ws + off; off += (size_t)N_NODES * DIM;   // h * norm
    float* bufC  = ws + off; off += (size_t)N_NODES * DIM;   // scatter acc
    float* x     = ws + off; off += (size_t)N_DEV * DIM;
    float* qkv   = ws + off; off += (size_t)N_DEV * 3 * DIM;
    float* attnb = ws + off; off += (size_t)N_DEV * DIM;
    float* tmpb  = ws + off; off += (size_t)N_DEV * DIM;
    float* qc    = ws + off; off += (size_t)N_DEV * DIM;
    float* kvc   = ws + off; off += (size_t)N_NODES * 2 * DIM;
    float* ffh   = ws + off; off += (size_t)N_DEV * FFDIM;
    float* cost  = ws + off; off += (size_t)(MROW + 1) * NCOL;
    float* costT = ws + off; off += (size_t)NCOL * (MROW + 1);
    float* uvec  = ws + off; off += 1056;
    float* vvec  = ws + off; off += NCOL;
    (void)in_sizes; (void)n_in; (void)out_size; (void)ws_size;

    const int B = 256;

    // ---- graph degree + norm -------------------------------------------
    zero_kernel<<<ceil_div(N_NODES, B), B, 0, stream>>>(deg, N_NODES);
    deg_kernel<<<ceil_div(N_EDGES, B), B, 0, stream>>>(edge_dst, deg, N_EDGES);
    norm_kernel<<<ceil_div(N_NODES, B), B, 0, stream>>>(deg, nrm, N_NODES);

    // ---- graph conv stack ----------------------------------------------
    const int  inF[4] = {32, 64, 64, 64};
    const int outF[4] = {64, 64, 64, 96};
    const float* h = node_feat;
    for (int l = 0; l < 4; ++l) {
        int F = inF[l], G = outF[l];
        int nEl = N_NODES * F;
        rowscale_kernel<<<ceil_div(nEl, B), B, 0, stream>>>(h, nrm, bufB, N_NODES, F);
        zero_kernel<<<ceil_div(nEl, B), B, 0, stream>>>(bufC, nEl);
        scatter_kernel<<<ceil_div(N_EDGES * F, B), B, 0, stream>>>(bufB, edge_src, edge_dst, bufC, F);
        combine_kernel<<<ceil_div(nEl, B), B, 0, stream>>>(bufC, bufB, nrm, N_NODES, F);
        launch_gemm(bufC, F, sg_w[l], F, sg_b[l], nullptr, bufA, G,
                    N_NODES, G, F, false, stream);
        h = bufA;
    }
    const float* cg = bufA;                       // 1024 x 96

    // ---- transformer ----------------------------------------------------
    copy_kernel<<<ceil_div(N_DEV * DIM, B), B, 0, stream>>>(device_feats, x, N_DEV * DIM);
    for (int l = 0; l < NLAYERS; ++l) {
        const float* sw  = sa_w  + (size_t)l * 3 * DIM * DIM;
        const float* sb  = sa_b  + (size_t)l * 3 * DIM;
        const float* sow = sa_ow + (size_t)l * DIM * DIM;
        const float* sob = sa_ob + (size_t)l * DIM;
        const float* cw  = ca_w  + (size_t)l * 3 * DIM * DIM;
        const float* cb  = ca_b  + (size_t)l * 3 * DIM;
        const float* cow = ca_ow + (size_t)l * DIM * DIM;
        const float* cob = ca_ob + (size_t)l * DIM;
        const float* w1  = ff_w1 + (size_t)l * FFDIM * DIM;
        const float* b1  = ff_b1 + (size_t)l * FFDIM;
        const float* w2  = ff_w2 + (size_t)l * DIM * FFDIM;
        const float* b2  = ff_b2 + (size_t)l * DIM;

        // self-attention
        launch_gemm(x, DIM, sw, DIM, sb, nullptr, qkv, 3 * DIM,
                    N_DEV, 3 * DIM, DIM, false, stream);
        attn_kernel<<<dim3(N_DEV / B, NHEAD), B, 0, stream>>>(
            qkv, 3 * DIM, qkv + DIM, 3 * DIM, qkv + 2 * DIM, 3 * DIM,
            attnb, DIM, N_DEV);
        launch_gemm(attnb, DIM, sow, DIM, sob, x, tmpb, DIM,
                    N_DEV, DIM, DIM, false, stream);
        ln_kernel<<<ceil_div(N_DEV, B), B, 0, stream>>>(
            tmpb, ln1_g + l * DIM, ln1_b + l * DIM, x, N_DEV, DIM);

        // cross-attention (queries from x, keys/values from cg)
        launch_gemm(x, DIM, cw, DIM, cb, nullptr, qc, DIM,
                    N_DEV, DIM, DIM, false, stream);
        launch_gemm(cg, DIM, cw + DIM * DIM, DIM, cb + DIM, nullptr, kvc, 2 * DIM,
                    N_NODES, 2 * DIM, DIM, false, stream);
        attn_kernel<<<dim3(N_DEV / B, NHEAD), B, 0, stream>>>(
            qc, DIM, kvc, 2 * DIM, kvc + DIM, 2 * DIM, attnb, DIM, N_NODES);
        launch_gemm(attnb, DIM, cow, DIM, cob, x, tmpb, DIM,
                    N_DEV, DIM, DIM, false, stream);
        ln_kernel<<<ceil_div(N_DEV, B), B, 0, stream>>>(
            tmpb, ln2_g + l * DIM, ln2_b + l * DIM, x, N_DEV, DIM);

        // feed-forward
        launch_gemm(x, DIM, w1, DIM, b1, nullptr, ffh, FFDIM,
                    N_DEV, FFDIM, DIM, true, stream);
        launch_gemm(ffh, FFDIM, w2, FFDIM, b2, x, tmpb, DIM,
                    N_DEV, DIM, FFDIM, false, stream);
        ln_kernel<<<ceil_div(N_DEV, B), B, 0, stream>>>(
            tmpb, ln3_g + l * DIM, ln3_b + l * DIM, x, N_DEV, DIM);
    }

    // ---- match head + cost matrix ---------------------------------------
    launch_gemm(x, DIM, match_w, DIM, match_b, nullptr, attnb, DIM,
                N_DEV, DIM, DIM, false, stream);           // attnb := match
    cost_kernel<<<ceil_div((MROW + 1) * NCOL, B), B, 0, stream>>>(cg, attnb, bin_sc,
                                                                  cost, costT);

    // ---- Sinkhorn --------------------------------------------------------
    zero_kernel<<<ceil_div(MROW + 1, B), B, 0, stream>>>(uvec, MROW + 1);
    zero_kernel<<<ceil_div(NCOL, B), B, 0, stream>>>(vvec, NCOL);
    for (int it = 0; it < SINK_IT; ++it) {
        sink_row_kernel<<<MROW + 1, 256, 0, stream>>>(cost, vvec, uvec);
        sink_col_kernel<<<NCOL, 256, 0, stream>>>(costT, uvec, vvec);
    }

    // ---- scores + greedy assignment -------------------------------------
    float* out = (float*)d_out;
    scores_kernel<<<ceil_div((MROW + 1) * NCOL, B), B, 0, stream>>>(cost, uvec, vvec, out);
    float* choices = out + (size_t)(MROW + 1) * NCOL;
    float* chsc    = choices + MROW;
    greedy_kernel<<<1, 1024, 0, stream>>>(out, choices, chsc);
}